// GCN_pyg_39986145525883
// MI455X (gfx1250) — compile-verified
//
#include <hip/hip_runtime.h>

#define NODES 379
#define NUM_GRAPHS 128
#define NTOT (NODES * NUM_GRAPHS)   // 48512 nodes
#define HID 128
#define IN_CH 379
#define NEDGE (NTOT * 32)           // 1552384 edges

typedef __attribute__((ext_vector_type(2))) float v2f;
typedef __attribute__((ext_vector_type(8))) float v8f;

__device__ __forceinline__ void atomAddF(float* p, float v) {
  __hip_atomic_fetch_add(p, v, __ATOMIC_RELAXED, __HIP_MEMORY_SCOPE_AGENT);
}

// ---------------------------------------------------------------- utilities
__global__ void fill_kernel(float* __restrict__ p, float v, int n) {
  int i = blockIdx.x * blockDim.x + threadIdx.x;
  if (i < n) p[i] = v;
}

__global__ void rsqrt_kernel(float* __restrict__ p, int n) {
  int i = blockIdx.x * blockDim.x + threadIdx.x;
  if (i < n) p[i] = rsqrtf(p[i]);
}

// ----------------------------------------------- per-edge weight + degree
__global__ void edge_weight_kernel(const int* __restrict__ src, const int* __restrict__ dst,
                                   const float* __restrict__ w, const float* __restrict__ lew,
                                   float* __restrict__ ew, float* __restrict__ deg, int E) {
  int e = blockIdx.x * blockDim.x + threadIdx.x;
  if (e >= E) return;
  int s = src[e], d = dst[e];
  int li = s % NODES, lj = d % NODES;
  float sm = 0.5f * (lew[li * NODES + lj] + lew[lj * NODES + li]);
  float le = 2.0f / (1.0f + __expf(-sm));
  float v = le * w[e];
  ew[e] = v;
  atomAddF(&deg[d], v);
}

// ------------------------------------------------- fp32 WMMA tiled GEMM
// C[M x 128] = A[M x K] * B[K x 128]; M = gridDim.x * 128 (M % 128 == 0).
// 256-thread block = 8 waves; wave w owns rows [w*16, w*16+16) x all 128 cols.
// B is staged in fragment-native layout: lds_b[(k>>2)*512 + col*4 + (k&3)],
// so a lane's two B values are one aligned ds_load_b64 and all 64 banks are
// hit conflict-free (lanes 0-15: banks 4*m16{+0,1}; lanes 16-31: +2,+3).
#define BM 128
#define BK 32
#define SA 36    // LDS A row stride (frag read: 16 distinct banks/half, disjoint halves)

__global__ __launch_bounds__(256) void gemm_wmma_f32(const float* __restrict__ A,
                                                     const float* __restrict__ B,
                                                     float* __restrict__ C, int K) {
  __shared__ float lds_a[BM * SA];
  __shared__ float lds_b[(BK / 4) * 512];   // 8 K-groups x 128 cols x 4
  const int tid  = threadIdx.x;
  const int wave = tid >> 5;
  const int lane = tid & 31;
  const int m16  = lane & 15;
  const int hf   = lane >> 4;              // half-wave id
  const int rowBase = blockIdx.x * BM;
  const int aBase = (wave * 16 + m16) * SA + hf * 2;

  v8f acc[8] = {};

  for (int k0 = 0; k0 < K; k0 += BK) {
    // stage A tile 128x32 (coalesced along K, zero-pad K tail)
    {
      int col = tid & 31;
      int r0  = tid >> 5;
      int k   = k0 + col;
      #pragma unroll
      for (int i = 0; i < 16; ++i) {
        int r = r0 + i * 8;
        lds_a[r * SA + col] = (k < K) ? A[(size_t)(rowBase + r) * K + k] : 0.0f;
      }
    }
    // stage B tile 32x128, swizzled to fragment-native groups of 4 K-values
    {
      int col = tid & 127;
      int r0  = tid >> 7;
      #pragma unroll
      for (int i = 0; i < 16; ++i) {
        int r = r0 + i * 2;
        int k = k0 + r;
        lds_b[(r >> 2) * 512 + col * 4 + (r & 3)] =
            (k < K) ? B[(size_t)k * HID + col] : 0.0f;
      }
    }
    __syncthreads();

    #pragma unroll
    for (int kk = 0; kk < BK; kk += 4) {
      // A frag: a[v] = A[row][kk + hf*2 + v]  (one b64 LDS load)
      v2f a = *(const v2f*)&lds_a[aBase + kk];
      const float* bg = &lds_b[(kk >> 2) * 512 + hf * 2];
      #pragma unroll
      for (int nt = 0; nt < 8; ++nt) {
        // B frag: b[v] = B[kk + hf*2 + v][nt*16 + m16]  (one b64 LDS load)
        v2f b = *(const v2f*)&bg[(nt * 16 + m16) * 4];
        acc[nt] = __builtin_amdgcn_wmma_f32_16x16x4_f32(
            false, a, false, b, (short)0, acc[nt], false, false);
      }
    }
    __syncthreads();
  }

  // D layout: vgpr r -> row r + hf*8; col = nt*16 + m16
  #pragma unroll
  for (int nt = 0; nt < 8; ++nt) {
    int col = nt * 16 + m16;
    #pragma unroll
    for (int r = 0; r < 8; ++r) {
      int row = rowBase + wave * 16 + r + hf * 8;
      C[(size_t)row * HID + col] = acc[nt][r];
    }
  }
}

// ----------------------------------------- edge scatter (one wave per edge)
// lane l handles channels [4l, 4l+4): coalesced 512B row gather, 4 f32 atomics.
__global__ void scatter_kernel(const float* __restrict__ xw, const float* __restrict__ ew,
                               const float* __restrict__ dinv, const int* __restrict__ src,
                               const int* __restrict__ dst, float* __restrict__ agg, int E) {
  int tid  = blockIdx.x * blockDim.x + threadIdx.x;
  int e    = tid >> 5;
  int lane = tid & 31;
  if (e >= E) return;
  int s = src[e], d = dst[e];
  float coeff = dinv[s] * ew[e] * dinv[d];
  const float4* xs = (const float4*)(xw + (size_t)s * HID);
  float4 v = xs[lane];
  float* ad = agg + (size_t)d * HID + lane * 4;
  atomAddF(ad + 0, coeff * v.x);
  atomAddF(ad + 1, coeff * v.y);
  atomAddF(ad + 2, coeff * v.z);
  atomAddF(ad + 3, coeff * v.w);
}

// --------------------------------------------------------- fused epilogues
__global__ void conv1_epilogue(const float* __restrict__ agg, const float* __restrict__ xw,
                               const float* __restrict__ xe, const float* __restrict__ dinv,
                               const float* __restrict__ b1, const float* __restrict__ be,
                               float* __restrict__ h, int n) {
  int t = blockIdx.x * blockDim.x + threadIdx.x;
  if (t >= n) return;
  int i = t >> 7, c = t & 127;
  float di = dinv[i];
  float t1 = agg[t] + di * di * xw[t] + b1[c];
  float t2 = xe[t] + be[c];
  h[t] = fmaxf(t1, 0.0f) + fmaxf(t2, 0.0f);
}

__global__ void conv2_epilogue(const float* __restrict__ agg, const float* __restrict__ hw,
                               const float* __restrict__ dinv, const float* __restrict__ b2,
                               float* __restrict__ h, int n) {
  int t = blockIdx.x * blockDim.x + threadIdx.x;
  if (t >= n) return;
  int i = t >> 7, c = t & 127;
  float di = dinv[i];
  h[t] += fmaxf(agg[t] + di * di * hw[t] + b2[c], 0.0f);
}

// ------------------------------------- mean-pool + FC fused (block = graph)
__global__ __launch_bounds__(256) void pool_fc_kernel(const float* __restrict__ h,
                                                      const float* __restrict__ Wfc,
                                                      const float* __restrict__ bfc,
                                                      float* __restrict__ out) {
  __shared__ float red[256];
  int g = blockIdx.x;
  const float* hg = h + (size_t)g * NODES * HID;
  float acc = 0.0f;
  for (int idx = threadIdx.x; idx < NODES * HID; idx += 256) {
    acc += hg[idx] * Wfc[idx & 127];
  }
  red[threadIdx.x] = acc;
  __syncthreads();
  for (int s = 128; s > 0; s >>= 1) {
    if (threadIdx.x < s) red[threadIdx.x] += red[threadIdx.x + s];
    __syncthreads();
  }
  if (threadIdx.x == 0) out[g] = red[0] / (float)NODES + bfc[0];
}

// --------------------------------------------------------------- launcher
extern "C" void kernel_launch(void* const* d_in, const int* in_sizes, int n_in,
                              void* d_out, int out_size, void* d_ws, size_t ws_size,
                              hipStream_t stream) {
  const float* x   = (const float*)d_in[0];
  const int*   ei  = (const int*)d_in[1];
  const float* w   = (const float*)d_in[2];
  // d_in[3] = batch (implicit: node i -> graph i/379)
  const float* lew = (const float*)d_in[4];
  const float* W1  = (const float*)d_in[5];
  const float* b1  = (const float*)d_in[6];
  const float* W2  = (const float*)d_in[7];
  const float* b2  = (const float*)d_in[8];
  const float* We  = (const float*)d_in[9];
  const float* be  = (const float*)d_in[10];
  const float* Wfc = (const float*)d_in[11];
  const float* bfc = (const float*)d_in[12];
  float* out = (float*)d_out;

  const int E = NEDGE;
  const int N = NTOT;
  const int NH = N * HID;
  const int* src = ei;
  const int* dst = ei + E;

  float* p   = (float*)d_ws;
  float* ew  = p;  p += E;
  float* deg = p;  p += N;                   // deg -> dinv in place
  float* bufA = p; p += (size_t)NH;          // xw1, later hw2
  float* bufB = p; p += (size_t)NH;          // xe,  later agg2
  float* bufC = p; p += (size_t)NH;          // agg1
  float* bufH = p; p += (size_t)NH;          // h, updated in place to h2

  dim3 blk(256);

  // degrees start at 1.0 (self-loop), accumulate effective edge weights
  fill_kernel<<<(N + 255) / 256, blk, 0, stream>>>(deg, 1.0f, N);
  edge_weight_kernel<<<(E + 255) / 256, blk, 0, stream>>>(src, dst, w, lew, ew, deg, E);
  rsqrt_kernel<<<(N + 255) / 256, blk, 0, stream>>>(deg, N);   // deg := deg^{-1/2}

  // xw1 = x@W1, xe = x@We  (fp32 WMMA, K=379 zero-padded)
  gemm_wmma_f32<<<N / BM, blk, 0, stream>>>(x, W1, bufA, IN_CH);
  gemm_wmma_f32<<<N / BM, blk, 0, stream>>>(x, We, bufB, IN_CH);

  // conv1 aggregation
  fill_kernel<<<(NH + 255) / 256, blk, 0, stream>>>(bufC, 0.0f, NH);
  scatter_kernel<<<(E * 32) / 256, blk, 0, stream>>>(bufA, ew, deg, src, dst, bufC, E);
  conv1_epilogue<<<(NH + 255) / 256, blk, 0, stream>>>(bufC, bufA, bufB, deg, b1, be, bufH, NH);

  // hw2 = h@W2 (K=128), conv2 aggregation, residual in place
  gemm_wmma_f32<<<N / BM, blk, 0, stream>>>(bufH, W2, bufA, HID);
  fill_kernel<<<(NH + 255) / 256, blk, 0, stream>>>(bufB, 0.0f, NH);
  scatter_kernel<<<(E * 32) / 256, blk, 0, stream>>>(bufA, ew, deg, src, dst, bufB, E);
  conv2_epilogue<<<(NH + 255) / 256, blk, 0, stream>>>(bufB, bufA, deg, b2, bufH, NH);

  // mean pool per graph + final FC
  pool_fc_kernel<<<NUM_GRAPHS, blk, 0, stream>>>(bufH, Wfc, bfc, out);
}